// Model_71330816852195
// MI455X (gfx1250) — compile-verified
//
#include <hip/hip_runtime.h>

// ---------------------------------------------------------------------------
// ESN forward, MI455X (gfx1250).   HBM-bound: reads x (25 MB) once.
// state_j = sum_t d_j^(S-1-t) * (M @ xs_t)_j,  M = W_in @ W_proj_in
//         = sum_k M[j,k] * G[j,k],  G[j,k] = sum_t d_j^(S-1-t) * xs[t,k]
// Block b: loads raw interleaved x[b] (both channels) contiguously via
// GLOBAL_LOAD_ASYNC_TO_LDS_B128 (double-buffered 64-segment phases), computes
// G partials for chunks (c=0,b) and (c=1,b) as W_d @ Xs with
// v_wmma_f32_16x16x4_f32; between phases acc is rescaled by d_j^64 (exact
// linear-recurrence combine). Chunk partials combined deterministically.
// ---------------------------------------------------------------------------

typedef __attribute__((ext_vector_type(2))) float v2f;
typedef __attribute__((ext_vector_type(8))) float v8f;

#define WIN 48            // window / reservoir size
#define HSEG 64           // segments per phase (K of each matmul phase)
#define NPART 1024        // chunk partials: c*512 + b, 128 segments each
#define GSZ (WIN * WIN)   // 2304
#define WD_STRIDE 68      // 68%64==4 -> b64 A-loads hit all 64 banks
#define PHFLOATS (HSEG * WIN * 2)   // 6144 floats per phase (both channels)

// ---------------------------------------------------------------------------
// Kernel 1: per-(b) block, 288 threads = 9 wave32s, one 16x16 tile each,
// two channel accumulators per wave. 64 WMMAs/wave.
// ---------------------------------------------------------------------------
__global__ __launch_bounds__(288) void esn_chunk_kernel(
    const float* __restrict__ x,   // (512, 6144, 2)
    const float* __restrict__ d,   // (48,)
    float* __restrict__ g_ws)      // (1024, 48, 48)
{
    __shared__ float sBuf[2][PHFLOATS];      // raw interleaved (l,c), 24 KB each
    __shared__ float sWd[WIN * WD_STRIDE];   // W_d[j][t] = d_j^(63-t), padded

    const int tid = threadIdx.x;
    const int b   = blockIdx.x;
    const float* xb = x + (size_t)b * 12288;

    // ---- issue async copy of phase-0 block (6144 floats, b128 per lane)
    for (int f4 = tid; f4 < PHFLOATS / 4; f4 += 288) {
        unsigned lva  = (unsigned)(size_t)&sBuf[0][f4 * 4];  // LDS byte address
        unsigned voff = (unsigned)(f4 * 16);
        asm volatile("global_load_async_to_lds_b128 %0, %1, %2"
                     :: "v"(lva), "v"(voff), "s"(xb) : "memory");
    }

    // ---- generate W_d while the copy is in flight (threads 0..47)
    if (tid < WIN) {
        float dj = d[tid];
        float p  = 1.0f;
        float* row = &sWd[tid * WD_STRIDE];
        for (int t = HSEG - 1; t >= 0; --t) { row[t] = p; p *= dj; }
    }

    asm volatile("s_wait_asynccnt 0x0" ::: "memory");
    __syncthreads();

    // ---- issue async copy of phase-1 block; overlapped with phase-0 WMMA
    for (int f4 = tid; f4 < PHFLOATS / 4; f4 += 288) {
        unsigned lva  = (unsigned)(size_t)&sBuf[1][f4 * 4];
        unsigned voff = (unsigned)(f4 * 16);
        asm volatile("global_load_async_to_lds_b128 %0, %1, %2"
                     :: "v"(lva), "v"(voff), "s"(xb + PHFLOATS) : "memory");
    }

    // ---- wave/tile setup (ISA 7.12.2 fragment layouts)
    const int wave = tid >> 5;
    const int lane = tid & 31;
    const int hf   = lane >> 4;      // lane-half selects K pair
    const int ml   = lane & 15;      // M (A) / N (B,C/D) index in tile
    const int m0   = (wave / 3) * 16;
    const int n0   = (wave % 3) * 16;

    v8f acc0 = {0.f, 0.f, 0.f, 0.f, 0.f, 0.f, 0.f, 0.f};
    v8f acc1 = acc0;
    const float* arow = &sWd[(m0 + ml) * WD_STRIDE];

    // ---- phase 0: G_p0 = W_d @ Xs[0:64]  (both channels)
    #pragma unroll 4
    for (int kk = 0; kk < HSEG / 4; ++kk) {
        const int kc = kk * 4 + 2 * hf;
        v2f a; a.x = arow[kc]; a.y = arow[kc + 1];
        const float2 p0 = *(const float2*)&sBuf[0][(kc * WIN + n0 + ml) * 2];
        const float2 p1 = *(const float2*)&sBuf[0][((kc + 1) * WIN + n0 + ml) * 2];
        v2f b0; b0.x = p0.x; b0.y = p1.x;    // channel 0
        v2f b1; b1.x = p0.y; b1.y = p1.y;    // channel 1
        acc0 = __builtin_amdgcn_wmma_f32_16x16x4_f32(false, a, false, b0,
                                                     (short)0, acc0, false, false);
        acc1 = __builtin_amdgcn_wmma_f32_16x16x4_f32(false, a, false, b1,
                                                     (short)0, acc1, false, false);
    }

    // ---- per-row rescale factors d_row^64 (row = m0 + r + 8*hf)
    float w64[8];
    #pragma unroll
    for (int r = 0; r < 8; ++r) {
        float w = d[m0 + r + 8 * hf];
        #pragma unroll
        for (int s = 0; s < 6; ++s) w *= w;
        w64[r] = w;
    }

    asm volatile("s_wait_asynccnt 0x0" ::: "memory");
    __syncthreads();

    // ---- G = d^64 * P0 + P1 : rescale then keep accumulating
    #pragma unroll
    for (int r = 0; r < 8; ++r) { acc0[r] *= w64[r]; acc1[r] *= w64[r]; }

    // ---- phase 1: accumulate W_d @ Xs[64:128]
    #pragma unroll 4
    for (int kk = 0; kk < HSEG / 4; ++kk) {
        const int kc = kk * 4 + 2 * hf;
        v2f a; a.x = arow[kc]; a.y = arow[kc + 1];
        const float2 p0 = *(const float2*)&sBuf[1][(kc * WIN + n0 + ml) * 2];
        const float2 p1 = *(const float2*)&sBuf[1][((kc + 1) * WIN + n0 + ml) * 2];
        v2f b0; b0.x = p0.x; b0.y = p1.x;
        v2f b1; b1.x = p0.y; b1.y = p1.y;
        acc0 = __builtin_amdgcn_wmma_f32_16x16x4_f32(false, a, false, b0,
                                                     (short)0, acc0, false, false);
        acc1 = __builtin_amdgcn_wmma_f32_16x16x4_f32(false, a, false, b1,
                                                     (short)0, acc1, false, false);
    }

    // ---- C/D layout: VGPR r -> row m0 + r + 8*hf, column n0 + ml
    float* g0 = g_ws + (size_t)b * GSZ;           // partial index b       (c=0)
    float* g1 = g_ws + (size_t)(512 + b) * GSZ;   // partial index 512+b   (c=1)
    #pragma unroll
    for (int r = 0; r < 8; ++r) {
        const int e = (m0 + r + 8 * hf) * WIN + (n0 + ml);
        g0[e] = acc0[r];
        g1[e] = acc1[r];
    }
}

// ---------------------------------------------------------------------------
// Kernel 2: deterministic chunk-level recurrence  G = d_j^128 * G + G_pi.
// 9 x 256 = 2304 threads, one (j,k) entry each; coalesced over k.
// ---------------------------------------------------------------------------
__global__ __launch_bounds__(256) void esn_combine_kernel(
    const float* __restrict__ d,
    const float* __restrict__ g_ws,   // (1024, 2304)
    float* __restrict__ g_tot)        // (2304,)
{
    const int e = blockIdx.x * 256 + threadIdx.x;
    if (e >= GSZ) return;
    const int j = e / WIN;

    float w = d[j];                    // d_j^128 via 7 squarings
    #pragma unroll
    for (int s = 0; s < 7; ++s) w *= w;

    float st = 0.0f;
    #pragma unroll 4
    for (int pi = 0; pi < NPART; ++pi)
        st = fmaf(w, st, g_ws[(size_t)pi * GSZ + e]);
    g_tot[e] = st;
}

// ---------------------------------------------------------------------------
// Kernel 3: fold M = W_in@W_proj_in, contract with G, project, scatter output.
// ---------------------------------------------------------------------------
__global__ __launch_bounds__(1024) void esn_final_kernel(
    const float* __restrict__ w_proj_in,   // (48,48) [p][k]
    const float* __restrict__ w_in,        // (48,48) [j][p]
    const float* __restrict__ w_proj_out,  // (48,48) [q][j]
    const float* __restrict__ g_tot,       // (48,48) [j][k]
    float* __restrict__ out)               // (512,3,2) flat = 3072
{
    __shared__ float sM[GSZ];
    __shared__ float sState[WIN];
    __shared__ float sY[WIN];
    const int tid = threadIdx.x;

    for (int e = tid; e < GSZ; e += 1024) {          // M = W_in @ W_proj_in
        const int j = e / WIN, k = e % WIN;
        float s = 0.0f;
        #pragma unroll 8
        for (int p = 0; p < WIN; ++p)
            s = fmaf(w_in[j * WIN + p], w_proj_in[p * WIN + k], s);
        sM[e] = s;
    }
    __syncthreads();

    if (tid < WIN) {                                 // state_j = <M[j,:], G[j,:]>
        float s = 0.0f;
        #pragma unroll 8
        for (int k = 0; k < WIN; ++k)
            s = fmaf(sM[tid * WIN + k], g_tot[tid * WIN + k], s);
        sState[tid] = s;
    }
    __syncthreads();

    if (tid < WIN) {                                 // y = W_proj_out @ state
        float s = 0.0f;
        #pragma unroll 8
        for (int j = 0; j < WIN; ++j)
            s = fmaf(w_proj_out[tid * WIN + j], sState[j], s);
        sY[tid] = s;
    }
    __syncthreads();

    // out flat (b,p,c): g = b*6 + p*2 + c ; value = y[(b*6 + c*3 + p) % 48]
    for (int g = tid; g < 3072; g += 1024) {
        const int b2 = g / 6, rem = g % 6;
        const int p = rem >> 1, c = rem & 1;
        out[g] = sY[(b2 * 6 + c * 3 + p) % WIN];
    }
}

// ---------------------------------------------------------------------------
extern "C" void kernel_launch(void* const* d_in, const int* in_sizes, int n_in,
                              void* d_out, int out_size, void* d_ws, size_t ws_size,
                              hipStream_t stream) {
    const float* x          = (const float*)d_in[0];  // (512,6144,2)
    const float* w_proj_in  = (const float*)d_in[1];  // (48,48)
    const float* w_in       = (const float*)d_in[2];  // (48,48)
    const float* d          = (const float*)d_in[3];  // (48,)
    const float* w_proj_out = (const float*)d_in[4];  // (48,48)
    float* out = (float*)d_out;

    float* g_ws  = (float*)d_ws;                       // 1024*2304 f32 = 9.44 MB
    float* g_tot = g_ws + (size_t)NPART * GSZ;         // +2304 f32

    esn_chunk_kernel<<<512, 288, 0, stream>>>(x, d, g_ws);
    esn_combine_kernel<<<9, 256, 0, stream>>>(d, g_ws, g_tot);
    esn_final_kernel<<<1, 1024, 0, stream>>>(w_proj_in, w_in, w_proj_out, g_tot, out);
}